// QLayer_45878840655942
// MI455X (gfx1250) — compile-verified
//
#include <hip/hip_runtime.h>

typedef float v2f __attribute__((ext_vector_type(2)));
typedef float v8f __attribute__((ext_vector_type(8)));

__device__ __forceinline__ float2 cmul(float2 a, float2 b) {
    return make_float2(a.x * b.x - a.y * b.y, a.x * b.y + a.y * b.x);
}
// a * conj(b)
__device__ __forceinline__ float2 cmulc(float2 a, float2 b) {
    return make_float2(a.x * b.x + a.y * b.y, a.y * b.x - a.x * b.y);
}
__device__ __forceinline__ float2 cadd(float2 a, float2 b) {
    return make_float2(a.x + b.x, a.y + b.y);
}

// One block = 256 threads = 8 waves; each wave computes phi for 32 batch rows
// via two 16x16x(8x4) f32 WMMA accumulations (only 3 output columns used),
// then every thread finishes one batch element (sincos + 2x2 complex sandwich).
__global__ __launch_bounds__(256) void qlayer_wmma_kernel(
    const float* __restrict__ rho_r, const float* __restrict__ rho_i,
    const float* __restrict__ x, const float* __restrict__ w,
    const float* __restrict__ theta, float* __restrict__ out)
{
    __shared__ float phiS[8][32][4];   // [wave][element][phi component], 4 KB

    const int tid  = threadIdx.x;
    const int lane = tid & 31;
    const int wv   = tid >> 5;
    const int n    = lane & 15;   // column / M-row index within a 16-group
    const int half = lane >> 4;   // 0: lanes 0-15, 1: lanes 16-31

    const long long base = (long long)blockIdx.x * 256;
    const long long g    = base + tid;          // this thread's batch element

    // Pull rho toward the caches while the matrix pipe works.
    __builtin_prefetch(rho_r + g * 4, 0, 0);
    __builtin_prefetch(rho_i + g * 4, 0, 0);

    // --- WMMA phase: phi[m][j] = sum_k x[m][k] * w[j][k] -------------------
    const long long r0    = base + (long long)wv * 32; // wave's first row
    const long long rowA0 = r0 + n;                    // tile0 row for lane
    const long long rowA1 = r0 + 16 + n;               // tile1 row for lane

    // B matrix = W^T (32x3) zero-padded to 32x16; mask without branching so
    // EXEC stays all-ones at the WMMAs.
    const int   wn    = (n < 3) ? n : 0;
    const float wmask = (n < 3) ? 1.0f : 0.0f;

    v8f acc0 = {};
    v8f acc1 = {};
    #pragma unroll
    for (int k = 0; k < 8; ++k) {
        const int kbase = 4 * k + 2 * half;  // per-ISA f32 A/B VGPR layout
        float2 a0 = *(const float2*)(x + rowA0 * 32 + kbase);
        float2 a1 = *(const float2*)(x + rowA1 * 32 + kbase);
        float2 bw = *(const float2*)(w + (long long)wn * 32 + kbase);
        v2f A0 = { a0.x, a0.y };
        v2f A1 = { a1.x, a1.y };
        v2f Bf = { bw.x * wmask, bw.y * wmask };
        acc0 = __builtin_amdgcn_wmma_f32_16x16x4_f32(
                   false, A0, false, Bf, (short)0, acc0, false, false);
        acc1 = __builtin_amdgcn_wmma_f32_16x16x4_f32(
                   false, A1, false, Bf, (short)0, acc1, false, false);
    }

    // Scatter the 3 useful output columns to LDS (accumulator VGPR index
    // depends on M, so this transpose needs memory, not shuffles).
    if (n < 3) {
        #pragma unroll
        for (int c = 0; c < 8; ++c) {
            const int m = c + 8 * half;       // C/D layout: VGPR c, half -> M
            phiS[wv][m][n]      = acc0[c];
            phiS[wv][16 + m][n] = acc1[c];
        }
    }
    __syncthreads();

    // --- Epilogue: one batch element per thread ----------------------------
    const float phi0 = phiS[wv][lane][0] + theta[0];
    const float phi1 = phiS[wv][lane][1] + theta[1];
    const float phi2 = phiS[wv][lane][2] + theta[2];

    const float h0 = 0.5f * phi0;
    const float a  = 0.5f * (phi1 + phi2);
    const float b  = 0.5f * (phi1 - phi2);
    float s, c, sa, ca, sb, cb;
    __sincosf(h0, &s,  &c);
    __sincosf(a,  &sa, &ca);
    __sincosf(b,  &sb, &cb);

    const float2 U00 = make_float2( c * ca,  c * sa);
    const float2 U01 = make_float2(-s * cb, -s * sb);
    const float2 U10 = make_float2( s * cb, -s * sb);
    const float2 U11 = make_float2( c * ca, -c * sa);

    const float4 rr = *(const float4*)(rho_r + g * 4);
    const float4 ri = *(const float4*)(rho_i + g * 4);
    const float2 p00 = make_float2(rr.x, ri.x);
    const float2 p01 = make_float2(rr.y, ri.y);
    const float2 p10 = make_float2(rr.z, ri.z);
    const float2 p11 = make_float2(rr.w, ri.w);

    // T = U * rho
    const float2 T00 = cadd(cmul(U00, p00), cmul(U01, p10));
    const float2 T01 = cadd(cmul(U00, p01), cmul(U01, p11));
    const float2 T10 = cadd(cmul(U10, p00), cmul(U11, p10));
    const float2 T11 = cadd(cmul(U10, p01), cmul(U11, p11));
    // R = T * U^dagger
    const float2 R00 = cadd(cmulc(T00, U00), cmulc(T01, U01));
    const float2 R01 = cadd(cmulc(T00, U10), cmulc(T01, U11));
    const float2 R10 = cadd(cmulc(T10, U00), cmulc(T11, U01));
    const float2 R11 = cadd(cmulc(T10, U10), cmulc(T11, U11));

    // complex64 interleaved (re, im), row-major (B,2,2)
    const float4 o0 = make_float4(R00.x, R00.y, R01.x, R01.y);
    const float4 o1 = make_float4(R10.x, R10.y, R11.x, R11.y);
    *(float4*)(out + g * 8)     = o0;
    *(float4*)(out + g * 8 + 4) = o1;
}

extern "C" void kernel_launch(void* const* d_in, const int* in_sizes, int n_in,
                              void* d_out, int out_size, void* d_ws, size_t ws_size,
                              hipStream_t stream) {
    const float* rho_r = (const float*)d_in[0];
    const float* rho_i = (const float*)d_in[1];
    const float* x     = (const float*)d_in[2];
    const float* w     = (const float*)d_in[3];
    const float* theta = (const float*)d_in[4];
    float* out = (float*)d_out;

    const int B = in_sizes[0] / 4;          // rho_r is (B,2,2)
    const int blocks = B / 256;             // B = 2^21 -> 8192 blocks, exact
    qlayer_wmma_kernel<<<blocks, 256, 0, stream>>>(rho_r, rho_i, x, w, theta, out);
}